// Decoder_4801773437026
// MI455X (gfx1250) — compile-verified
//
#include <hip/hip_runtime.h>
#include <hip/hip_bf16.h>
#include <math.h>

// Problem constants (from reference): B=128, T=256, H=512
#define BB 128
#define TT 256
#define HH 512

typedef float v2f __attribute__((ext_vector_type(2)));
typedef float v8f __attribute__((ext_vector_type(8)));

// ---------------------------------------------------------------------------
// f32 WMMA GEMM: C[M,N] = A[M,K] * B[K,N] (+ bias[N] if given).
// One wave32 per 16x32 C tile (two 16x16 accumulators sharing the A fragment),
// K advanced 4 per V_WMMA_F32_16X16X4_F32.  LDB is a compile-time constant so
// the B-operand addresses strength-reduce to constant-offset loads (no per-
// iteration v_mul_u64).
//
// Fragment layouts per CDNA5 ISA 7.12.2:
//   A (16x4 f32, 2 VGPR): lanes 0-15 -> M=lane, {K=k0,k0+1}; lanes 16-31 -> {K=k0+2,k0+3}
//   B (4x16 f32, 2 VGPR): mirrored (row striped across lanes within a VGPR)
//   C/D (16x16 f32, 8 VGPR): VGPR j: lanes 0-15 = row j, lanes 16-31 = row j+8
// ---------------------------------------------------------------------------
template <int LDB>
__global__ void wmma_gemm_f32(const float* __restrict__ A,
                              const float* __restrict__ B,
                              const float* __restrict__ bias,
                              float* __restrict__ C,
                              int M, int N, int K,
                              int lda, int ldc)
{
    const int wave = (blockIdx.x * blockDim.x + threadIdx.x) >> 5;
    const int lane = threadIdx.x & 31;
    const int ntiles = N >> 5;                 // 32-wide N tiles
    const int m0 = (wave / ntiles) << 4;
    const int n0 = (wave % ntiles) << 5;
    if (m0 >= M) return;

    const int half = lane >> 4;     // 0: lanes 0-15 (K pair 0,1); 1: lanes 16-31 (K pair 2,3)
    const int l    = lane & 15;

    // A fragment: contiguous pair -> single b64 load per lane per iteration
    const float* Ap = A + (size_t)(m0 + l) * lda + half * 2;
    // B fragments: two 16-wide column groups, rows kb and kb+1
    const float* Bp = B + (size_t)(half * 2) * LDB + n0 + l;

    v8f acc0 = {};
    v8f acc1 = {};
    const int iters = K >> 2;
#pragma unroll 4
    for (int it = 0; it < iters; ++it) {
        v2f a = *(const v2f*)Ap;               // A[m][kb], A[m][kb+1]
        v2f b0, b1;
        b0.x = Bp[0];        b0.y = Bp[LDB];
        b1.x = Bp[16];       b1.y = Bp[LDB + 16];
        // pull the streaming B panel into near caches ahead of time
        __builtin_prefetch(Bp + 16 * LDB, 0, 3);
        acc0 = __builtin_amdgcn_wmma_f32_16x16x4_f32(
            false, a, false, b0, (short)0, acc0, false, false);
        acc1 = __builtin_amdgcn_wmma_f32_16x16x4_f32(
            false, a, false, b1, (short)0, acc1, false, false);
        Ap += 4;
        Bp += (size_t)4 * LDB;
    }

    const float bv0 = bias ? bias[n0 + l]      : 0.0f;
    const float bv1 = bias ? bias[n0 + 16 + l] : 0.0f;
    float* Crow = C + (size_t)(m0 + half * 8) * ldc + n0 + l;
#pragma unroll
    for (int j = 0; j < 8; ++j) {
        Crow[(size_t)j * ldc]      = acc0[j] + bv0;
        Crow[(size_t)j * ldc + 16] = acc1[j] + bv1;
    }
}

// ---------------------------------------------------------------------------
// Build Wcomb = [Wk ; Wr]  ([768, 2048]) so z = [lp|h] @ Wcomb is one GEMM.
// ---------------------------------------------------------------------------
__global__ void build_wcomb(const float* __restrict__ Wk,
                            const float* __restrict__ Wr,
                            float* __restrict__ Wcomb)
{
    const size_t i = (size_t)blockIdx.x * blockDim.x + threadIdx.x;
    const size_t total = (size_t)(TT + HH) * (4 * HH);
    if (i >= total) return;
    const size_t r = i / (4 * HH);
    Wcomb[i] = (r < TT) ? Wk[i] : Wr[i - (size_t)TT * (4 * HH)];
}

// ---------------------------------------------------------------------------
// Init state: A[:,0:256] = 1 (lp0), A[:,256:768] = h0 ; c = c0.
// ---------------------------------------------------------------------------
__global__ void init_state(const float* __restrict__ h0,
                           const float* __restrict__ c0,
                           float* __restrict__ Abuf,
                           float* __restrict__ cbuf)
{
    const int i = blockIdx.x * blockDim.x + threadIdx.x;
    if (i < BB * (TT + HH)) {
        const int col = i % (TT + HH);
        const int b   = i / (TT + HH);
        Abuf[i] = (col < TT) ? 1.0f : h0[b * HH + (col - TT)];
    }
    if (i < BB * HH) cbuf[i] = c0[i];
}

// ---------------------------------------------------------------------------
// LSTM elementwise: gates z=[i|f|g|o] ([128,2048]) -> c (in place),
// h_new written into A columns 256..767.
// ---------------------------------------------------------------------------
__global__ void lstm_elem(const float* __restrict__ z,
                          float* __restrict__ c,
                          float* __restrict__ Abuf)
{
    const int i = blockIdx.x * blockDim.x + threadIdx.x;  // 0 .. 128*512-1
    if (i >= BB * HH) return;
    const int b = i >> 9;
    const int h = i & (HH - 1);
    const float* zr = z + (size_t)b * (4 * HH);
    const float zi = zr[h];
    const float zf = zr[HH + h];
    const float zg = zr[2 * HH + h];
    const float zo = zr[3 * HH + h];
    const float si = 1.0f / (1.0f + __expf(-zi));
    const float sf = 1.0f / (1.0f + __expf(-zf));
    const float so = 1.0f / (1.0f + __expf(-zo));
    const float cn = sf * c[i] + si * tanhf(zg);
    c[i] = cn;
    Abuf[(size_t)b * (TT + HH) + TT + h] = so * tanhf(cn);
}

// ---------------------------------------------------------------------------
// Attention + softmax, one block (256 threads = 8 waves) per batch row b.
//   score[b,t] = sum_h tanh(w1e[b,t,h] + w2d[b,h]) * V[h]
//   probs = softmax_t(score); write probs into A[:,0:256] and out[b,step,:].
// Each lane owns 16 contiguous h (float4 loads, fully coalesced from L2 —
// w1e (64MB) is L2-resident on MI455X's 192MB L2).
// ---------------------------------------------------------------------------
__global__ void attn_softmax(const float* __restrict__ w1e,
                             const float* __restrict__ w2d,
                             const float* __restrict__ V,
                             float* __restrict__ Abuf,
                             float* __restrict__ out,
                             int step)
{
    __shared__ float s_wd[HH];
    __shared__ float s_v[HH];
    __shared__ float s_score[TT];
    __shared__ float s_red[8];

    const int b    = blockIdx.x;
    const int tid  = threadIdx.x;          // 0..255
    const int wv   = tid >> 5;             // wave 0..7
    const int lane = tid & 31;

    s_wd[tid]       = w2d[b * HH + tid];
    s_wd[tid + 256] = w2d[b * HH + tid + 256];
    s_v[tid]        = V[tid];
    s_v[tid + 256]  = V[tid + 256];
    __syncthreads();

    // 8 waves x 32 t-values each
    for (int t = wv; t < TT; t += 8) {
        const float* row = w1e + ((size_t)b * TT + t) * HH + lane * 16;
        float acc = 0.0f;
#pragma unroll
        for (int j = 0; j < 16; j += 4) {
            const float4 e = *(const float4*)(row + j);
            const int h = lane * 16 + j;
            acc += tanhf(e.x + s_wd[h + 0]) * s_v[h + 0];
            acc += tanhf(e.y + s_wd[h + 1]) * s_v[h + 1];
            acc += tanhf(e.z + s_wd[h + 2]) * s_v[h + 2];
            acc += tanhf(e.w + s_wd[h + 3]) * s_v[h + 3];
        }
#pragma unroll
        for (int m = 16; m > 0; m >>= 1) acc += __shfl_xor(acc, m, 32);
        if (lane == 0) s_score[t] = acc;
    }
    __syncthreads();

    // block softmax over 256 scores
    const float s = s_score[tid];
    float mx = s;
#pragma unroll
    for (int m = 16; m > 0; m >>= 1) mx = fmaxf(mx, __shfl_xor(mx, m, 32));
    if (lane == 0) s_red[wv] = mx;
    __syncthreads();
    float gmax = s_red[0];
#pragma unroll
    for (int i = 1; i < 8; ++i) gmax = fmaxf(gmax, s_red[i]);
    __syncthreads();

    const float e = __expf(s - gmax);
    float sum = e;
#pragma unroll
    for (int m = 16; m > 0; m >>= 1) sum += __shfl_xor(sum, m, 32);
    if (lane == 0) s_red[wv] = sum;
    __syncthreads();
    float gsum = 0.0f;
#pragma unroll
    for (int i = 0; i < 8; ++i) gsum += s_red[i];

    const float p = e / gsum;
    Abuf[(size_t)b * (TT + HH) + tid] = p;                       // feeds next step's lp
    out[((size_t)b * TT + step) * TT + tid] = p;                 // [B, step, T]
}

// ---------------------------------------------------------------------------
extern "C" void kernel_launch(void* const* d_in, const int* in_sizes, int n_in,
                              void* d_out, int out_size, void* d_ws, size_t ws_size,
                              hipStream_t stream)
{
    const float* enc  = (const float*)d_in[0];  // [128,256,512]
    const float* h0   = (const float*)d_in[1];  // [128,512]
    const float* c0   = (const float*)d_in[2];  // [128,512]
    const float* W1   = (const float*)d_in[3];  // [512,512]
    const float* W2   = (const float*)d_in[4];  // [512,512]
    const float* Vv   = (const float*)d_in[5];  // [512]
    const float* Wk   = (const float*)d_in[6];  // [256,2048]
    const float* Wr   = (const float*)d_in[7];  // [512,2048]
    const float* bias = (const float*)d_in[8];  // [2048]
    float* out = (float*)d_out;                 // [128,256,256]

    // workspace carve (floats)
    float* ws    = (float*)d_ws;
    float* Wcomb = ws;                                   // 768*2048
    float* w1e   = Wcomb + (size_t)(TT + HH) * (4 * HH); // 32768*512
    float* Abuf  = w1e + (size_t)BB * TT * HH;           // 128*768  ([lp|h])
    float* cbuf  = Abuf + (size_t)BB * (TT + HH);        // 128*512
    float* zbuf  = cbuf + (size_t)BB * HH;               // 128*2048
    float* w2d   = zbuf + (size_t)BB * 4 * HH;           // 128*512

    const int BLK = 256;  // 8 waves per block on wave32

    // one-time setup
    build_wcomb<<<((TT + HH) * 4 * HH + BLK - 1) / BLK, BLK, 0, stream>>>(Wk, Wr, Wcomb);
    init_state<<<(BB * (TT + HH) + BLK - 1) / BLK, BLK, 0, stream>>>(h0, c0, Abuf, cbuf);

    // w1e = enc @ W1 : M=32768, N=512, K=512 -> 2048 x 16 (16x32) tiles = 32768 waves
    {
        const int tiles = (BB * TT / 16) * (HH / 32);
        wmma_gemm_f32<HH><<<tiles / 8, BLK, 0, stream>>>(enc, W1, nullptr, w1e,
                                                         BB * TT, HH, HH, HH, HH);
    }

    // 256 sequential decode steps
    for (int step = 0; step < TT; ++step) {
        // z = [lp|h] @ Wcomb + b : M=128, N=2048, K=768 -> 8 x 64 tiles = 512 waves
        wmma_gemm_f32<4 * HH><<<(8 * 64) / 8, BLK, 0, stream>>>(Abuf, Wcomb, bias, zbuf,
                                                                BB, 4 * HH, TT + HH,
                                                                TT + HH, 4 * HH);
        // LSTM elementwise -> c, h (into Abuf cols 256..767)
        lstm_elem<<<(BB * HH) / BLK, BLK, 0, stream>>>(zbuf, cbuf, Abuf);

        // w2d = h @ W2 : M=128, N=512, K=512 -> 8 x 16 tiles = 128 waves
        wmma_gemm_f32<HH><<<(8 * 16) / 8, BLK, 0, stream>>>(Abuf + TT, W2, nullptr, w2d,
                                                            BB, HH, HH,
                                                            TT + HH, HH);
        // attention + softmax -> probs into Abuf cols 0..255 and out[:,step,:]
        attn_softmax<<<BB, BLK, 0, stream>>>(w1e, w2d, Vv, Abuf, out, step);
    }
}